// FLPGNN_EdgeAttr_53506702573932
// MI455X (gfx1250) — compile-verified
//
#include <hip/hip_runtime.h>

#define NN 10000   // N_NODES
#define HID 64

typedef __attribute__((ext_vector_type(2))) float v2f;
typedef __attribute__((ext_vector_type(8))) float v8f;

// ---------------- utility ----------------
__global__ void zero_kernel(float* __restrict__ p, int n) {
  int i = blockIdx.x * blockDim.x + threadIdx.x;
  if (i < n) p[i] = 0.0f;
}

__global__ void count_kernel(const int* __restrict__ ei, float* __restrict__ cnt, int E) {
  int e = blockIdx.x * blockDim.x + threadIdx.x;
  if (e < E) atomicAdd(&cnt[ei[E + e]], 1.0f);
}

// ---------------- layer 1: per-edge message + scatter-add ----------------
// msg1[e,o] = sum_k relu(a*w1[k]+b1[k]) * (x0*w2[k,o]+x1*w2[k,64+o]+x2*w2[k,128+o])
//           + x0*b2[o]+x1*b2[64+o]+x2*b2[128+o]
__global__ void layer1_edge(const float* __restrict__ x, const int* __restrict__ ei,
                            const float* __restrict__ ea,
                            const float* __restrict__ w1, const float* __restrict__ b1,
                            const float* __restrict__ w2, const float* __restrict__ b2,
                            float* __restrict__ sum1, int E) {
  int o = threadIdx.x;                       // 0..63 (output channel)
  int e = blockIdx.x * blockDim.y + threadIdx.y;
  if (e >= E) return;
  int src = ei[e], dst = ei[E + e];
  float a  = ea[e];
  float x0 = x[src * 3 + 0], x1 = x[src * 3 + 1], x2 = x[src * 3 + 2];
  float acc = x0 * b2[o] + x1 * b2[64 + o] + x2 * b2[128 + o];
#pragma unroll 4
  for (int k = 0; k < 32; ++k) {
    float rh = fmaxf(a * w1[k] + b1[k], 0.0f);
    const float* wr = w2 + k * 192;          // e1_w2 row [192] = (i,o) flattened
    acc += rh * (x0 * wr[o] + x1 * wr[64 + o] + x2 * wr[128 + o]);
  }
  atomicAdd(&sum1[dst * HID + o], acc);
}

// h1 = relu(mean1 + x@root1 + bias1)
__global__ void combine1(const float* __restrict__ x, const float* __restrict__ sum1,
                         const float* __restrict__ cnt, const float* __restrict__ root1,
                         const float* __restrict__ bias1, float* __restrict__ h1) {
  int o = threadIdx.x;
  int n = blockIdx.x * blockDim.y + threadIdx.y;
  if (n >= NN) return;
  float inv = 1.0f / fmaxf(cnt[n], 1.0f);
  float v = sum1[n * HID + o] * inv
          + x[n * 3 + 0] * root1[o] + x[n * 3 + 1] * root1[64 + o] + x[n * 3 + 2] * root1[128 + o]
          + bias1[o];
  h1[n * HID + o] = fmaxf(v, 0.0f);
}

// ---------------- generic sliced GEMM on v_wmma_f32_16x16x4_f32 ----------------
// C[node, slice, col] = sum_i A[node, i] * B[slice][i, col]   (A: [N,64], B slice: [64,64] row-major)
// grid: (N/16 row-tiles, 64/16 col-tiles, S slices); block: 32 threads (one wave).
__global__ void gemm_slices_wmma(const float* __restrict__ A, const float* __restrict__ B,
                                 float* __restrict__ C, int bSliceStride,
                                 int cNodeStride, int cSliceStride) {
  int lane    = threadIdx.x;              // 0..31
  int rowTile = blockIdx.x;
  int colTile = blockIdx.y;
  int slice   = blockIdx.z;
  int l15   = lane & 15;
  int khalf = (lane >> 4) * 2;            // lanes 0-15 -> K {0,1}; lanes 16-31 -> K {2,3}
  int row   = rowTile * 16 + l15;         // A-matrix M = lane%16 in both halves
  int col   = colTile * 16 + l15;         // B-matrix N = lane%16 in both halves
  const float* Bs = B + (size_t)slice * bSliceStride;
  v8f c = {};
#pragma unroll
  for (int ks = 0; ks < 16; ++ks) {       // K = 64 in steps of 4
    int k0 = ks * 4 + khalf;
    v2f a, b;
    a.x = A[row * HID + k0];
    a.y = A[row * HID + k0 + 1];
    b.x = Bs[k0 * HID + col];
    b.y = Bs[(k0 + 1) * HID + col];
    // D = A(16x4) x B(4x16) + C ; 8-arg form (neg_a, A, neg_b, B, c_mod, C, reuse_a, reuse_b)
    c = __builtin_amdgcn_wmma_f32_16x16x4_f32(false, a, false, b, (short)0, c, false, false);
  }
  int m0 = (lane >> 4) * 8;               // C/D: vgpr v -> M = v + 8*(lane>=16), N = lane%16
#pragma unroll
  for (int v = 0; v < 8; ++v) {
    int r = rowTile * 16 + m0 + v;
    C[(size_t)r * cNodeStride + (size_t)slice * cSliceStride + colTile * 16 + l15] = c[v];
  }
}

// ---------------- layer 2: per-edge contraction against node-precomputed P2 ----------------
// msg2[e,o] = sum_k relu(a*w1[k]+b1[k]) * P2[src, k, o] + Q2[src, o]
__global__ void layer2_edge(const int* __restrict__ ei, const float* __restrict__ ea,
                            const float* __restrict__ w1, const float* __restrict__ b1,
                            const float* __restrict__ P2, const float* __restrict__ Q2,
                            float* __restrict__ sum2, int E) {
  __shared__ float rh[4][32];
  int o  = threadIdx.x;                   // 0..63
  int ly = threadIdx.y;                   // 0..3 edges per block
  int e  = blockIdx.x * 4 + ly;
  bool valid = (e < E);
  int src = 0, dst = 0;
  float a = 0.0f;
  if (valid) { src = ei[e]; dst = ei[E + e]; a = ea[e]; }
  if (o < 32) rh[ly][o] = fmaxf(a * w1[o] + b1[o], 0.0f);
  __syncthreads();
  if (!valid) return;
  const float* P = P2 + (size_t)src * (32 * HID);
  float acc = Q2[src * HID + o];
#pragma unroll 8
  for (int k = 0; k < 32; ++k) acc += rh[ly][k] * P[k * HID + o];
  atomicAdd(&sum2[dst * HID + o], acc);
}

// h2 = relu(mean2 + R2 + bias2);  out[n] = h2 . lin_w + lin_b
__global__ void combine2_out(const float* __restrict__ sum2, const float* __restrict__ cnt,
                             const float* __restrict__ R2, const float* __restrict__ bias2,
                             const float* __restrict__ lin_w, const float* __restrict__ lin_b,
                             float* __restrict__ out) {
  __shared__ float red[4][HID];
  int o  = threadIdx.x;
  int ly = threadIdx.y;
  int n  = blockIdx.x * 4 + ly;
  float partial = 0.0f;
  if (n < NN) {
    float inv = 1.0f / fmaxf(cnt[n], 1.0f);
    float h = sum2[n * HID + o] * inv + R2[n * HID + o] + bias2[o];
    partial = fmaxf(h, 0.0f) * lin_w[o];
  }
  red[ly][o] = partial;
  __syncthreads();
  if (o < 32) red[ly][o] += red[ly][o + 32];
  __syncthreads();
  if (o < 16) red[ly][o] += red[ly][o + 16];
  __syncthreads();
  if (o == 0 && n < NN) {
    float s = 0.0f;
#pragma unroll
    for (int i = 0; i < 16; ++i) s += red[ly][i];
    out[n] = s + lin_b[0];
  }
}

extern "C" void kernel_launch(void* const* d_in, const int* in_sizes, int n_in,
                              void* d_out, int out_size, void* d_ws, size_t ws_size,
                              hipStream_t stream) {
  const float* x     = (const float*)d_in[0];
  const int*   ei    = (const int*)  d_in[1];
  const float* ea    = (const float*)d_in[2];
  const float* e1w1  = (const float*)d_in[3];
  const float* e1b1  = (const float*)d_in[4];
  const float* e1w2  = (const float*)d_in[5];
  const float* e1b2  = (const float*)d_in[6];
  const float* root1 = (const float*)d_in[7];
  const float* bias1 = (const float*)d_in[8];
  const float* e2w1  = (const float*)d_in[9];
  const float* e2b1  = (const float*)d_in[10];
  const float* e2w2  = (const float*)d_in[11];   // [32,4096] = [k][i*64+o]
  const float* e2b2  = (const float*)d_in[12];   // [4096]    = [i*64+o]
  const float* root2 = (const float*)d_in[13];   // [64,64]
  const float* bias2 = (const float*)d_in[14];
  const float* lin_w = (const float*)d_in[15];
  const float* lin_b = (const float*)d_in[16];
  float* out = (float*)d_out;

  const int E = in_sizes[2];                     // edge_attr flat count (ea_dim = 1)

  // workspace layout (fp32): cnt[N] | sumb[N*64] | h1[N*64] | Q2[N*64] | R2[N*64] | P2[N*32*64]
  float* ws   = (float*)d_ws;
  float* cnt  = ws;
  float* sumb = cnt  + NN;           // shared by layer-1 and layer-2 scatter sums
  float* h1   = sumb + (size_t)NN * HID;
  float* Q2   = h1   + (size_t)NN * HID;
  float* R2   = Q2   + (size_t)NN * HID;
  float* P2   = R2   + (size_t)NN * HID;

  // ---- init + degree count (reused for both layers) ----
  zero_kernel<<<(NN + 255) / 256, 256, 0, stream>>>(cnt, NN);
  zero_kernel<<<(NN * HID + 255) / 256, 256, 0, stream>>>(sumb, NN * HID);
  count_kernel<<<(E + 255) / 256, 256, 0, stream>>>(ei, cnt, E);

  // ---- layer 1 ----
  layer1_edge<<<(E + 3) / 4, dim3(64, 4), 0, stream>>>(x, ei, ea, e1w1, e1b1, e1w2, e1b2, sumb, E);
  combine1<<<NN / 4, dim3(64, 4), 0, stream>>>(x, sumb, cnt, root1, bias1, h1);

  // ---- WMMA GEMMs: P2 = h1 x w2-slices, Q2 = h1 x b2, R2 = h1 x root2 ----
  gemm_slices_wmma<<<dim3(NN / 16, HID / 16, 32), 32, 0, stream>>>(h1, e2w2, P2, 4096, 32 * HID, HID);
  gemm_slices_wmma<<<dim3(NN / 16, HID / 16, 1),  32, 0, stream>>>(h1, e2b2, Q2, 0, HID, 0);
  gemm_slices_wmma<<<dim3(NN / 16, HID / 16, 1),  32, 0, stream>>>(h1, root2, R2, 0, HID, 0);

  // ---- layer 2 ----
  zero_kernel<<<(NN * HID + 255) / 256, 256, 0, stream>>>(sumb, NN * HID);
  layer2_edge<<<(E + 3) / 4, dim3(64, 4), 0, stream>>>(ei, ea, e2w1, e2b1, P2, Q2, sumb, E);
  combine2_out<<<NN / 4, dim3(64, 4), 0, stream>>>(sumb, cnt, R2, bias2, lin_w, lin_b, out);
}